// LearnableFilterbank_45947560133290
// MI455X (gfx1250) — compile-verified
//
#include <hip/hip_runtime.h>
#include <hip/hip_bf16.h>
#include <math.h>
#include <stdint.h>

typedef __attribute__((ext_vector_type(2))) float v2f;
typedef __attribute__((ext_vector_type(8))) float v8f;
typedef __attribute__((ext_vector_type(4))) unsigned int v4u;
typedef __attribute__((ext_vector_type(8))) int v8i;
typedef __attribute__((ext_vector_type(4))) int v4i;

#define NUM_FILTERS 128
#define KBINS 1025          // N_FFT/2 + 1
#define KCHUNK 64           // TDM chunk width (dwords / k-values)
#define NCHUNK 17
#define KPAD  (KCHUNK * NCHUNK)   // 1088, zero-padded
#define TDIM  2000
#define BDIM  32
#define LDS_ROW 66          // 64 data dwords + 2 pad dwords (bank skew)
#define CHUNK_DWORDS (NUM_FILTERS * LDS_ROW)   // 8448 dwords / 33792 B
#define WPB 16              // waves per block (512 threads)

// ---------------------------------------------------------------------------
// Kernel 1: build the learnable filterbank, zero-padded to KPAD cols.
// ---------------------------------------------------------------------------
__global__ void build_filterbank_kernel(const float* __restrict__ center_freqs,
                                        const float* __restrict__ bandwidths,
                                        const float* __restrict__ filter_shapes,
                                        float* __restrict__ fb /* [128][KPAD] */) {
    const int f = blockIdx.x;
    const float center = center_freqs[f];
    const float bw = logf(1.0f + expf(bandwidths[f])) + 0.001f;
    const float s0 = logf(1.0f + expf(filter_shapes[2 * f + 0])) + 0.1f;
    const float s1 = logf(1.0f + expf(filter_shapes[2 * f + 1])) + 0.1f;
    const float left  = center - bw * s0;
    const float right = center + bw * s1;

    for (int k = threadIdx.x; k < KPAD; k += blockDim.x) {
        float resp = 0.0f;
        if (k < KBINS) {
            const float fr = (float)k;
            const bool rise = (fr >= left) && (fr <= center);
            const bool fall = (fr > center) && (fr <= right);
            if (rise)      resp = (fr - left)   / (center - left  + 1e-8f);
            else if (fall) resp = 1.0f - (fr - center) / (right - center + 1e-8f);
        }
        fb[(size_t)f * KPAD + k] = resp;
    }
}

// ---------------------------------------------------------------------------
// TDM descriptor (D#) build + issue: 2D tile, 64 cols x 128 rows of f32,
// row-padded in LDS by 2 dwords (pad_interval=64 dwords, pad_amount=2 dwords)
// so A-reads (ds_load_b64, one row per lane) hit distinct banks.
// Bitfields per CDNA5 ISA.
// This toolchain's builtin is the 6-arg form:
//   (uint32x4 g0, int32x8 g1, int32x4 g2, int32x4 g3, int32x8 extra, i32 cpol)
// ---------------------------------------------------------------------------
__device__ __forceinline__ void tdm_load_fb_chunk(const float* gsrc, unsigned lds_byte_off) {
    const unsigned long long ga = (unsigned long long)(uintptr_t)gsrc;
    v4u g0;
    g0.x = 1u;                                   // count=1 (valid user descriptor)
    g0.y = lds_byte_off;                         // lds_addr
    g0.z = (unsigned)(ga & 0xFFFFFFFFu);         // global_addr[31:0]
    g0.w = (unsigned)((ga >> 32) & 0x01FFFFFFu)  // global_addr[56:32]
         | (2u << 30);                           // type = 2 ("image")
    v8i g1;
    g1[0] = (int)((2u << 16)                     // data_size = 4 bytes
                | (1u << 20)                     // pad_enable
                | (5u << 22)                     // pad_interval: 64 dwords
                | (1u << 25));                   // pad_amount: 2 dwords
    const unsigned td0 = KPAD, td1 = NUM_FILTERS;
    g1[1] = (int)((td0 & 0xFFFFu) << 16);                              // tensor_dim0 lo16
    g1[2] = (int)(((td0 >> 16) & 0xFFFFu) | ((td1 & 0xFFFFu) << 16));  // td0 hi | td1 lo
    g1[3] = (int)(((td1 >> 16) & 0xFFFFu) | ((unsigned)KCHUNK << 16)); // td1 hi | tile_dim0
    g1[4] = (int)NUM_FILTERS;                                          // tile_dim1 | tile_dim2=0
    g1[5] = (int)KPAD;                                                 // tensor_dim0_stride lo32
    g1[6] = 0;                                                         // stride0 hi | stride1 lo
    g1[7] = 0;                                                         // stride1 hi
    const v4i z4 = {0, 0, 0, 0};                 // groups 2/3 unused (2D tensor)
    const v8i z8 = {0, 0, 0, 0, 0, 0, 0, 0};
    __builtin_amdgcn_tensor_load_to_lds(g0, g1, z4, z4, z8, 0);
}

// ---------------------------------------------------------------------------
// Kernel 2: fp32 WMMA GEMM with TDM-staged filterbank.
//   filtered[b,f,t] = sum_k FB[f,k] * spec[b,k,t]
// One wave32 per (b, 16-wide t-tile); all 8 f-tiles per wave (64 acc VGPRs).
// FB K-chunks are DMA'd to LDS (double buffered) by wave 0 and shared by all
// 16 waves of the block; spectrum streams HBM->VGPR exactly once.
// ---------------------------------------------------------------------------
__global__ __launch_bounds__(WPB * 32) void mel_wmma_kernel(
        const float* __restrict__ spec,   // [32][1025][2000]
        const float* __restrict__ fb,     // [128][KPAD]
        float* __restrict__ out) {        // [32][2000][128]
    __shared__ float abuf[2 * CHUNK_DWORDS];

    const int lane = threadIdx.x & 31;
    const int wv   = threadIdx.x >> 5;
    const int wid  = blockIdx.x * WPB + wv;                 // 0..3999
    const int b    = wid / 125;
    const int t0   = (wid % 125) * 16;
    const int hi   = lane >> 4;                             // K-pair select
    const int ln   = lane & 15;                             // M for A, N for B/C

    const unsigned lds_base = (unsigned)(uintptr_t)(void*)abuf;

    v8f acc[8] = {};

    const float* __restrict__ specb = spec + (size_t)b * KBINS * TDIM + (t0 + ln);

    // Prologue: prefetch chunks 0 and 1 via the Tensor Data Mover.
    if (wv == 0) {
        tdm_load_fb_chunk(fb,          lds_base);
        tdm_load_fb_chunk(fb + KCHUNK, lds_base + CHUNK_DWORDS * 4u);
    }

    for (int c = 0; c < NCHUNK; ++c) {
        if (wv == 0) {
            if (c + 1 < NCHUNK) __builtin_amdgcn_s_wait_tensorcnt(1);
            else                __builtin_amdgcn_s_wait_tensorcnt(0);
        }
        __syncthreads();                                    // buffer (c&1) is ready

        const float* __restrict__ ab = abuf + (c & 1) * CHUNK_DWORDS;
        const int kc = c * KCHUNK;

        if (kc + KCHUNK <= KBINS) {
            // Fully in-bounds chunks (c = 0..15).
            for (int kl = 0; kl < KCHUNK; kl += 4) {
                const int kg = kc + kl + 2 * hi;            // global k of this lane's pair
                v2f bv;
                bv.x = specb[(size_t)(kg + 0) * TDIM];
                bv.y = specb[(size_t)(kg + 1) * TDIM];
#pragma unroll
                for (int ft = 0; ft < 8; ++ft) {
                    const v2f av = *(const v2f*)(ab + (ft * 16 + ln) * LDS_ROW + kl + 2 * hi);
                    acc[ft] = __builtin_amdgcn_wmma_f32_16x16x4_f32(
                        false, av, false, bv, (short)0, acc[ft], false, false);
                }
            }
        } else {
            // Tail chunk (c = 16, k = 1024..1087): FB pad cols are zero; guard spec.
            for (int kl = 0; kl < KCHUNK; kl += 4) {
                const int kg = kc + kl + 2 * hi;
                v2f bv;
                bv.x = (kg + 0 < KBINS) ? specb[(size_t)(kg + 0) * TDIM] : 0.0f;
                bv.y = (kg + 1 < KBINS) ? specb[(size_t)(kg + 1) * TDIM] : 0.0f;
#pragma unroll
                for (int ft = 0; ft < 8; ++ft) {
                    const v2f av = *(const v2f*)(ab + (ft * 16 + ln) * LDS_ROW + kl + 2 * hi);
                    acc[ft] = __builtin_amdgcn_wmma_f32_16x16x4_f32(
                        false, av, false, bv, (short)0, acc[ft], false, false);
                }
            }
        }

        __syncthreads();                                    // all waves done with (c&1)
        if (wv == 0 && c + 2 < NCHUNK) {
            tdm_load_fb_chunk(fb + (size_t)(c + 2) * KCHUNK,
                              lds_base + (unsigned)(c & 1) * (CHUNK_DWORDS * 4u));
        }
    }

    // Epilogue: reference reshape(F,B,T).transpose(1,2,0) maps to
    //   out[f & 31, t, 4*b + (f >> 5)] = filtered[b, f, t]
#pragma unroll
    for (int ft = 0; ft < 8; ++ft) {
        const int f0      = ft * 16;
        const int col     = 4 * b + (f0 >> 5);
        const int rowbase = (f0 & 31);
#pragma unroll
        for (int r = 0; r < 8; ++r) {
            const int m = r + 8 * hi;
            const size_t idx = (size_t)(rowbase + m) * (TDIM * NUM_FILTERS)
                             + (size_t)(t0 + ln) * NUM_FILTERS + col;
            out[idx] = acc[ft][r];
        }
    }
}

// ---------------------------------------------------------------------------
extern "C" void kernel_launch(void* const* d_in, const int* in_sizes, int n_in,
                              void* d_out, int out_size, void* d_ws, size_t ws_size,
                              hipStream_t stream) {
    const float* spectrum      = (const float*)d_in[0];  // (32, 1025, 2000) f32
    const float* center_freqs  = (const float*)d_in[1];  // (128,) f32
    const float* bandwidths    = (const float*)d_in[2];  // (128,) f32
    const float* filter_shapes = (const float*)d_in[3];  // (128, 2) f32

    float* fb = (float*)d_ws;                            // 128*1088*4 = 557 KB scratch
    build_filterbank_kernel<<<NUM_FILTERS, 256, 0, stream>>>(
        center_freqs, bandwidths, filter_shapes, fb);

    // 4000 waves (32 batches x 125 t-tiles), 16 waves per 512-thread block.
    mel_wmma_kernel<<<4000 / WPB, WPB * 32, 0, stream>>>(spectrum, fb, (float*)d_out);
}